// EntropicBinLoss_68453188764036
// MI455X (gfx1250) — compile-verified
//
#include <hip/hip_runtime.h>

#define EPS      1e-8f
#define NTOT     16384
#define FDIM     64
#define KDIM     32
#define FK       2048      // FDIM*KDIM
#define CDIM     100
#define CPAD     112       // 7 tiles of 16; column 100 = virtual ones column -> bin_mass
#define NSPLIT   16
#define NCHUNK   (NTOT / NSPLIT)

typedef float v2f __attribute__((ext_vector_type(2)));
typedef float v8f __attribute__((ext_vector_type(8)));

// -------------------------------------------------------------------------
// Kernel 1: S[fk, c] = sum_n membership[n, fk] * teacherPad[n, c]
//   teacherPad[n, c] = teacher[n,c] (c<100), 1.0 (c==100), 0.0 (c>100)
//   => S[:,100] == bin_mass (without EPS)
// One wave handles a 16-row fk tile across all 7 c-tiles.
// A-frag (16x4 f32): lanes 0-15: M=lane, K=0/1 in v0/v1; lanes 16-31: K=2/3.
// B-frag (4x16 f32): lanes 0-15: N=lane, K=0/1; lanes 16-31: K=2/3.
// -------------------------------------------------------------------------
__global__ __launch_bounds__(128) void ebl_gemm(const float* __restrict__ M,
                                                const float* __restrict__ T,
                                                float* __restrict__ S) {
  const int lane = threadIdx.x & 31;
  const int wave = threadIdx.x >> 5;
  const int m0   = (blockIdx.x * 4 + wave) * 16;       // fk tile base
  const int n_begin = blockIdx.y * NCHUNK;
  const int n_end   = n_begin + NCHUNK;

  const int row = lane & 15;                 // M (for A) / N (for B) within tile
  const int kk  = (lane >> 4) * 2;           // K slot base: 0 (low lanes) / 2 (high)

  v8f acc[7];
#pragma unroll
  for (int t = 0; t < 7; ++t) acc[t] = v8f{0.f,0.f,0.f,0.f,0.f,0.f,0.f,0.f};

  for (int n0 = n_begin; n0 < n_end; n0 += 4) {
    // ---- A fragment: membership rows n0+kk, n0+kk+1; columns m0+row ----
    v2f a;
    a.x = M[(size_t)(n0 + kk)     * FK + (m0 + row)];
    a.y = M[(size_t)(n0 + kk + 1) * FK + (m0 + row)];

#pragma unroll
    for (int t = 0; t < 7; ++t) {
      const int c  = t * 16 + row;
      const int cc = (c < CDIM) ? c : (CDIM - 1);      // clamp: keep loads in-bounds
      float bx = T[(size_t)(n0 + kk)     * CDIM + cc];
      float by = T[(size_t)(n0 + kk + 1) * CDIM + cc];
      if (c >= CDIM) {                                  // pad region: ones col @ c==100
        const float pad = (c == CDIM) ? 1.0f : 0.0f;
        bx = pad;
        by = pad;
      }
      v2f b; b.x = bx; b.y = by;
      // D = A(16x4) * B(4x16) + C  -- exact f32 WMMA
      acc[t] = __builtin_amdgcn_wmma_f32_16x16x4_f32(
          /*neg_a=*/false, a, /*neg_b=*/false, b,
          /*c_mod=*/(short)0, acc[t], /*reuse_a=*/false, /*reuse_b=*/false);
    }
  }

  // D layout (32-bit 16x16): VGPR r -> lanes 0-15: M=r, N=lane; lanes 16-31: M=r+8.
  const int mbase = m0 + (lane >> 4) * 8;
#pragma unroll
  for (int t = 0; t < 7; ++t) {
    const int c = t * 16 + row;
#pragma unroll
    for (int r = 0; r < 8; ++r) {
      atomicAdd(&S[(size_t)(mbase + r) * CPAD + c], acc[t][r]);
    }
  }
}

// -------------------------------------------------------------------------
// Kernel 2: entropy epilogue over S[2048, 112] (mass in column 100).
//   loss_intra = -(1/N) * sum_{fk,c} S * log(S/mass + EPS)
//   loss_inter = sum_{f,k<K-1,c} mix*log(mix+EPS), mix = 0.5*(cent_k + cent_{k+1})
//   out = loss_intra + 0.5 * loss_inter
// -------------------------------------------------------------------------
__global__ __launch_bounds__(256) void ebl_loss(const float* __restrict__ S,
                                                float* __restrict__ out) {
  const int f = blockIdx.x;                  // 0..63
  const float invN = 1.0f / (float)NTOT;
  float local = 0.0f;

  // intra-bin term
  for (int idx = threadIdx.x; idx < KDIM * CDIM; idx += 256) {
    const int k = idx / CDIM, c = idx % CDIM;
    const size_t base = (size_t)(f * KDIM + k) * CPAD;
    const float s    = S[base + c];
    const float mass = S[base + CDIM] + EPS;
    const float cent = s / mass;
    local -= s * __logf(cent + EPS) * invN;
  }
  // inter-bin (adjacent mixture) term
  for (int idx = threadIdx.x; idx < (KDIM - 1) * CDIM; idx += 256) {
    const int k = idx / CDIM, c = idx % CDIM;
    const size_t b0 = (size_t)(f * KDIM + k) * CPAD;
    const size_t b1 = b0 + CPAD;
    const float c0 = S[b0 + c] / (S[b0 + CDIM] + EPS);
    const float c1 = S[b1 + c] / (S[b1 + CDIM] + EPS);
    const float mix = 0.5f * (c0 + c1);
    local += 0.5f * mix * __logf(mix + EPS);
  }

  __shared__ float red[256];
  red[threadIdx.x] = local;
  __syncthreads();
  for (int s = 128; s > 0; s >>= 1) {
    if (threadIdx.x < s) red[threadIdx.x] += red[threadIdx.x + s];
    __syncthreads();
  }
  if (threadIdx.x == 0) atomicAdd(out, red[0]);
}

// -------------------------------------------------------------------------
extern "C" void kernel_launch(void* const* d_in, const int* in_sizes, int n_in,
                              void* d_out, int out_size, void* d_ws, size_t ws_size,
                              hipStream_t stream) {
  const float* membership = (const float*)d_in[0];   // [N, F, K] = [16384, 2048]
  const float* teacher    = (const float*)d_in[1];   // [N, C]    = [16384, 100]
  float* out = (float*)d_out;                        // scalar f32
  float* S   = (float*)d_ws;                         // [2048, 112] accumulator

  hipMemsetAsync(S, 0, (size_t)FK * CPAD * sizeof(float), stream);
  hipMemsetAsync(out, 0, sizeof(float), stream);

  dim3 grid1(FK / (16 * 4), NSPLIT);                 // 32 x 16 blocks, 4 waves each
  ebl_gemm<<<grid1, 128, 0, stream>>>(membership, teacher, S);
  ebl_loss<<<FDIM, 256, 0, stream>>>(S, out);
}